// GAT_Block_65566970741156
// MI455X (gfx1250) — compile-verified
//
#include <hip/hip_runtime.h>

// ---------------------------------------------------------------------------
// GAT block for MI455X (gfx1250, wave32).
//   h      = x @ W                      (fp32 WMMA 16x16x4, 16x128 per wave)
//   s_src  = h @ a[:128], s_dst = h @ a[128:]
//   attn_e = leaky_relu(s_src[src_e] + s_dst[dst_e], 0.2)
//   out[dst] += attn_e * h[src]        (hardware f32 atomics, L2-resident)
// ---------------------------------------------------------------------------

typedef float v2f __attribute__((ext_vector_type(2)));
typedef float v8f __attribute__((ext_vector_type(8)));

#define C_IN   256
#define C_OUT  128
#define ALPHA  0.2f

// ---------------------------------------------------------------------------
// Zero the output buffer (harness poisons d_out with 0xAA before timing).
// ---------------------------------------------------------------------------
__global__ void gat_zero_kernel(float* __restrict__ p, long long n4) {
    long long i      = (long long)blockIdx.x * blockDim.x + threadIdx.x;
    long long stride = (long long)gridDim.x * blockDim.x;
    float4 z = make_float4(0.f, 0.f, 0.f, 0.f);
    for (; i < n4; i += stride) ((float4*)p)[i] = z;
}

// ---------------------------------------------------------------------------
// h = x @ W via V_WMMA_F32_16X16X4_F32.
// One wave computes a 16x128 row-strip: 8 accumulator fragments (64 VGPRs,
// spill-free), K loop of 64 steps (4 per WMMA). A-fragment: contiguous
// float2 (global_load_b64); B fragments clause-batched b32 loads.
// ---------------------------------------------------------------------------
__global__ void gat_gemm_kernel(const float* __restrict__ x,
                                const float* __restrict__ W,
                                float* __restrict__ h,
                                int n_nodes, int n_tiles) {
    const int lane = threadIdx.x & 31;
    const int wave = threadIdx.x >> 5;                 // 8 waves / block
    const int tile = blockIdx.x * 8 + wave;            // 16-row tile index
    if (tile >= n_tiles) return;                       // wave-uniform exit

    const int m0   = tile * 16;
    const int half = lane >> 4;                        // 0: lanes 0-15, 1: 16-31
    const int lmod = lane & 15;

    // Clamp load row (keeps EXEC all-ones for WMMA even on a ragged tail).
    int row = m0 + lmod;
    if (row > n_nodes - 1) row = n_nodes - 1;
    const float* __restrict__ xrow = x + (long long)row * C_IN;

    v8f acc[8] = {};                                   // 16x128 fp32 tile

    for (int k0 = 0; k0 < C_IN; k0 += 4) {
        // A fragment: lane holds x[row][k0+2*half .. k0+2*half+1]
        v2f a;
        a.x = xrow[k0 + 2 * half + 0];
        a.y = xrow[k0 + 2 * half + 1];
#pragma unroll
        for (int nt = 0; nt < 8; ++nt) {
            const int col = nt * 16 + lmod;
            v2f b;                                     // B: K rows striped over halves
            b.x = W[(long long)(k0 + 2 * half + 0) * C_OUT + col];
            b.y = W[(long long)(k0 + 2 * half + 1) * C_OUT + col];
            acc[nt] = __builtin_amdgcn_wmma_f32_16x16x4_f32(
                /*neg_a=*/false, a, /*neg_b=*/false, b,
                /*c_mod=*/(short)0, acc[nt],
                /*reuse_a=*/false, /*reuse_b=*/false);
        }
    }

    // D layout: VGPR r, lane l -> element (M = r + 8*half, N = lmod + 16*nt).
    // Fast path (always taken for N % 16 == 0): unguarded coalesced stores,
    // no exec-mask churn. Ragged path kept only for generality.
    if (m0 + 16 <= n_nodes) {
#pragma unroll
        for (int nt = 0; nt < 8; ++nt)
#pragma unroll
            for (int r = 0; r < 8; ++r)
                h[(long long)(m0 + r + 8 * half) * C_OUT + nt * 16 + lmod] = acc[nt][r];
    } else {
#pragma unroll
        for (int nt = 0; nt < 8; ++nt)
#pragma unroll
            for (int r = 0; r < 8; ++r) {
                const int m = m0 + r + 8 * half;
                if (m < n_nodes)
                    h[(long long)m * C_OUT + nt * 16 + lmod] = acc[nt][r];
            }
    }
}

// ---------------------------------------------------------------------------
// s_src[i] = dot(h[i], a[0:128]) ; s_dst[i] = dot(h[i], a[128:256])
// One wave per node: lane = 4 channels, shuffle-tree reduction.
// ---------------------------------------------------------------------------
__global__ void gat_scores_kernel(const float* __restrict__ h,
                                  const float* __restrict__ a,
                                  float* __restrict__ s_src,
                                  float* __restrict__ s_dst,
                                  int n_nodes) {
    const int lane = threadIdx.x & 31;
    const int node = (int)(((long long)blockIdx.x * blockDim.x + threadIdx.x) >> 5);
    if (node >= n_nodes) return;

    const float4 hv = ((const float4*)(h + (long long)node * C_OUT))[lane];
    const float4 w1 = ((const float4*)(a))[lane];
    const float4 w2 = ((const float4*)(a + C_OUT))[lane];

    float p1 = hv.x * w1.x + hv.y * w1.y + hv.z * w1.z + hv.w * w1.w;
    float p2 = hv.x * w2.x + hv.y * w2.y + hv.z * w2.z + hv.w * w2.w;

#pragma unroll
    for (int off = 16; off > 0; off >>= 1) {
        p1 += __shfl_xor(p1, off, 32);
        p2 += __shfl_xor(p2, off, 32);
    }
    if (lane == 0) { s_src[node] = p1; s_dst[node] = p2; }
}

// ---------------------------------------------------------------------------
// Edge scatter: one wave per edge, lane = 4 channels.
// h (51 MB) and out (51 MB) both fit in the 192 MB L2, so the random
// gather + f32 atomic adds resolve at L2 bandwidth, not HBM.
// ---------------------------------------------------------------------------
__global__ void gat_edge_kernel(const int* __restrict__ ei,
                                const float* __restrict__ h,
                                const float* __restrict__ s_src,
                                const float* __restrict__ s_dst,
                                float* __restrict__ out,
                                int n_edges) {
    const int lane    = threadIdx.x & 31;
    const int wid     = (int)(((long long)blockIdx.x * blockDim.x + threadIdx.x) >> 5);
    const int n_waves = (int)(((long long)gridDim.x * blockDim.x) >> 5);

    for (int e = wid; e < n_edges; e += n_waves) {
        const int src = ei[e];
        const int dst = ei[n_edges + e];

        float sc = s_src[src] + s_dst[dst];
        sc = (sc > 0.f) ? sc : ALPHA * sc;             // leaky relu

        const float4 hv = ((const float4*)(h + (long long)src * C_OUT))[lane];
        float* o = out + (long long)dst * C_OUT + lane * 4;
        unsafeAtomicAdd(o + 0, sc * hv.x);             // global_atomic_add_f32
        unsafeAtomicAdd(o + 1, sc * hv.y);
        unsafeAtomicAdd(o + 2, sc * hv.z);
        unsafeAtomicAdd(o + 3, sc * hv.w);
    }
}

// ---------------------------------------------------------------------------
// Launch
// ---------------------------------------------------------------------------
extern "C" void kernel_launch(void* const* d_in, const int* in_sizes, int n_in,
                              void* d_out, int out_size, void* d_ws, size_t ws_size,
                              hipStream_t stream) {
    const float* x  = (const float*)d_in[0];   // [N, 256]
    const int*   ei = (const int*)  d_in[1];   // [2, E]
    const float* W  = (const float*)d_in[2];   // [256, 128]
    const float* a  = (const float*)d_in[3];   // [256]
    float* out = (float*)d_out;                // [N, 128]

    const int N = in_sizes[0] / C_IN;          // 100000
    const int E = in_sizes[1] / 2;             // 1600000

    // Workspace: h [N,128] fp32, then s_src [N], s_dst [N]
    float* h     = (float*)d_ws;
    float* s_src = h + (long long)N * C_OUT;
    float* s_dst = s_src + N;

    // 1) zero output (atomics accumulate into it)
    {
        long long n4 = (long long)N * C_OUT / 4;
        gat_zero_kernel<<<2048, 256, 0, stream>>>(out, n4);
    }

    // 2) h = x @ W   (fp32 WMMA; 8 waves/block, one 16x128 strip per wave)
    {
        const int n_tiles = (N + 15) / 16;     // 6250
        const int blocks  = (n_tiles + 7) / 8; // 782
        gat_gemm_kernel<<<blocks, 256, 0, stream>>>(x, W, h, N, n_tiles);
    }

    // 3) attention score projections (one wave per node)
    {
        const int blocks = (int)(((long long)N * 32 + 255) / 256);  // 12500
        gat_scores_kernel<<<blocks, 256, 0, stream>>>(h, a, s_src, s_dst, N);
    }

    // 4) edge gather/scale/scatter-add (one wave per edge, grid-stride)
    {
        gat_edge_kernel<<<4096, 256, 0, stream>>>(ei, h, s_src, s_dst, out, E);
    }
}